// PointerNet_18056042512704
// MI455X (gfx1250) — compile-verified
//
#include <hip/hip_runtime.h>
#include <math.h>

// Problem constants from the reference.
#define B_  32
#define S_  128
#define T_  64
#define K_  512   // E == Q == 512
#define H_  256
#define TQ  4     // t-values fused per block in stage 2

typedef __attribute__((ext_vector_type(2))) float v2f;
typedef __attribute__((ext_vector_type(8))) float v8f;

#if __has_builtin(__builtin_amdgcn_tanhf)
#define FAST_TANH(x) __builtin_amdgcn_tanhf(x)
#elif __has_builtin(__builtin_amdgcn_tanh_f32)
#define FAST_TANH(x) __builtin_amdgcn_tanh_f32(x)
#else
#define FAST_TANH(x) tanhf(x)
#endif

// ---------------------------------------------------------------------------
// C[M,N] = A[M,K] @ W[N,K]^T + bias[N], fp32, via V_WMMA_F32_16X16X4_F32.
// One 16x16 C tile per wave; 8 waves / block.
//  A (16x4 f32):  lane l -> row M = l&15, VGPR v -> K = (l>>4)*2 + v
//  B (4x16 f32):  lane l -> col N = l&15, VGPR v -> K = (l>>4)*2 + v
//  C (16x16 f32): VGPR r -> row (r + (l>>4)*8), col = l&15
// ---------------------------------------------------------------------------
__global__ __launch_bounds__(256) void gemm_xt_bias_wmma(
    const float* __restrict__ A,    // [M, K] row-major
    const float* __restrict__ W,    // [N, K] row-major (nn.Linear weight)
    const float* __restrict__ bias, // [N]
    float* __restrict__ C,          // [M, N]
    int M, int N, int K)
{
  const int lane  = threadIdx.x & 31;
  const int wave  = threadIdx.x >> 5;
  const int tilesN = N >> 4;
  const int tile  = blockIdx.x * 8 + wave;
  const int tm    = (tile / tilesN) << 4;
  const int tn    = (tile % tilesN) << 4;
  if (tm >= M) return;              // wave-uniform: EXEC stays all-ones for WMMA

  const int half = lane >> 4;       // selects K sub-pair {0,1} vs {2,3}
  const int l16  = lane & 15;

  const float* arow = A + (size_t)(tm + l16) * K + (half << 1);
  const float* wrow = W + (size_t)(tn + l16) * K + (half << 1);

  v8f acc = {};
  #pragma unroll 8
  for (int k = 0; k < K; k += 4) {
    v2f a = *(const v2f*)(arow + k);   // 8B, 8B-aligned
    v2f b = *(const v2f*)(wrow + k);
    acc = __builtin_amdgcn_wmma_f32_16x16x4_f32(
        /*neg_a=*/false, a, /*neg_b=*/false, b,
        /*c_mod=*/(short)0, acc, /*reuse_a=*/false, /*reuse_b=*/false);
  }

  const float bv = bias[tn + l16];
  float* crow = C + (size_t)(tm + (half << 3)) * N + tn + l16;
  #pragma unroll
  for (int r = 0; r < 8; ++r)
    crow[(size_t)r * N] = acc[r] + bv;
}

// ---------------------------------------------------------------------------
// Fused: h2[t,b,s] = sum_h w2[h]*tanh(src_proj[b,s,h] + q_proj[t,b,h]) + b2
//        -> mask -> softmax over s.
// One block = one b and TQ consecutive t. 128 threads = 4 waves.
// Lanes stripe H (h = lane + 32j); each wave owns s = wave, wave+4, ...
// ---------------------------------------------------------------------------
__global__ __launch_bounds__(128) void attn_tanh_softmax(
    const float* __restrict__ src_proj, // [B*S, H]
    const float* __restrict__ q_proj,   // [T*B, H]
    const float* __restrict__ w2,       // [H]
    const float* __restrict__ b2p,      // [1]
    const int*  __restrict__ mask,      // [B*S] (bool as int32)
    float* __restrict__ out)            // [T*B, S]
{
  const int b    = blockIdx.x & (B_ - 1);
  const int t0   = (blockIdx.x >> 5) * TQ;   // B_ == 32
  const int lane = threadIdx.x & 31;
  const int wave = threadIdx.x >> 5;

  __shared__ float h2s[TQ][S_];
  __shared__ float redmax[TQ][4];
  __shared__ float redsum[TQ][4];

  float wv[8], qv[TQ][8];
  #pragma unroll
  for (int j = 0; j < 8; ++j) wv[j] = w2[lane + 32 * j];
  #pragma unroll
  for (int tt = 0; tt < TQ; ++tt) {
    const float* qrow = q_proj + (size_t)((t0 + tt) * B_ + b) * H_;
    #pragma unroll
    for (int j = 0; j < 8; ++j) qv[tt][j] = qrow[lane + 32 * j];
  }
  const float bias2 = b2p[0];

  for (int s = wave; s < S_; s += 4) {
    const float* srow = src_proj + (size_t)(b * S_ + s) * H_;
    float x[8];
    #pragma unroll
    for (int j = 0; j < 8; ++j) x[j] = srow[lane + 32 * j];
    #pragma unroll
    for (int tt = 0; tt < TQ; ++tt) {
      float acc = 0.0f;
      #pragma unroll
      for (int j = 0; j < 8; ++j)
        acc = fmaf(wv[j], FAST_TANH(x[j] + qv[tt][j]), acc);
      #pragma unroll
      for (int off = 16; off; off >>= 1)     // wave32 reduction
        acc += __shfl_xor(acc, off);
      if (lane == 0) h2s[tt][s] = acc + bias2;
    }
  }
  __syncthreads();

  // Softmax over S=128: thread i owns s=i.
  const int s = threadIdx.x;
  const bool masked = (mask[b * S_ + s] != 0);
  float vv[TQ], ev[TQ];
  #pragma unroll
  for (int tt = 0; tt < TQ; ++tt) {
    vv[tt] = masked ? -INFINITY : h2s[tt][s];
    float m = vv[tt];
    #pragma unroll
    for (int off = 16; off; off >>= 1) m = fmaxf(m, __shfl_xor(m, off));
    if (lane == 0) redmax[tt][wave] = m;
  }
  __syncthreads();
  #pragma unroll
  for (int tt = 0; tt < TQ; ++tt) {
    const float m = fmaxf(fmaxf(redmax[tt][0], redmax[tt][1]),
                          fmaxf(redmax[tt][2], redmax[tt][3]));
    ev[tt] = __expf(vv[tt] - m);
    float sum = ev[tt];
    #pragma unroll
    for (int off = 16; off; off >>= 1) sum += __shfl_xor(sum, off);
    if (lane == 0) redsum[tt][wave] = sum;
  }
  __syncthreads();
  #pragma unroll
  for (int tt = 0; tt < TQ; ++tt) {
    const float sum = redsum[tt][0] + redsum[tt][1] +
                      redsum[tt][2] + redsum[tt][3];
    out[(size_t)((t0 + tt) * B_ + b) * S_ + s] = ev[tt] / sum;
  }
}

extern "C" void kernel_launch(void* const* d_in, const int* in_sizes, int n_in,
                              void* d_out, int out_size, void* d_ws, size_t ws_size,
                              hipStream_t stream) {
  (void)in_sizes; (void)n_in; (void)out_size; (void)ws_size;
  const float* src_enc = (const float*)d_in[0];  // (B,S,E)
  const int*   mask    = (const int*)  d_in[1];  // (B,S) bool->int
  const float* query   = (const float*)d_in[2];  // (T,B,Q)
  const float* W_src   = (const float*)d_in[3];  // (H,E)
  const float* b_src   = (const float*)d_in[4];  // (H)
  const float* W_q     = (const float*)d_in[5];  // (H,Q)
  const float* b_q     = (const float*)d_in[6];  // (H)
  const float* w2      = (const float*)d_in[7];  // (H)
  const float* b2      = (const float*)d_in[8];  // ()
  float* out = (float*)d_out;                    // (T,B,S)

  float* src_proj = (float*)d_ws;                        // 4096*256 f32 = 4 MB
  float* q_proj   = src_proj + (size_t)(B_ * S_) * H_;   // 2048*256 f32 = 2 MB

  // src_proj: M=4096, N=256, K=512 -> (256*16)/8 = 512 blocks
  gemm_xt_bias_wmma<<<512, 256, 0, stream>>>(src_enc, W_src, b_src, src_proj,
                                             B_ * S_, H_, K_);
  // q_proj: M=2048 -> (128*16)/8 = 256 blocks
  gemm_xt_bias_wmma<<<256, 256, 0, stream>>>(query, W_q, b_q, q_proj,
                                             T_ * B_, H_, K_);
  // fused tanh-dot-softmax: B * T/TQ = 512 blocks
  attn_tanh_softmax<<<B_ * (T_ / TQ), 128, 0, stream>>>(src_proj, q_proj, w2,
                                                        b2, mask, out);
}